// SetDSR_6820408066672
// MI455X (gfx1250) — compile-verified
//
#include <hip/hip_runtime.h>

#define EPSF 1e-8f
#define NEG_INFF (-1e9f)

typedef __attribute__((ext_vector_type(2))) float v2f;
typedef __attribute__((ext_vector_type(8))) float v8f;

// Hardware transcendental helpers (v_exp_f32 = 2^x, v_log_f32 = log2(x),
// v_sin/v_cos take revolutions -> pre-scale by 1/(2*pi)).
__device__ __forceinline__ float fexp(float x) {
  return __builtin_amdgcn_exp2f(x * 1.4426950408889634f);
}
__device__ __forceinline__ float flog(float x) {
  return __builtin_amdgcn_logf(x) * 0.6931471805599453f;
}
__device__ __forceinline__ float fsin(float x) {
  return __builtin_amdgcn_sinf(x * 0.15915494309189535f);
}
__device__ __forceinline__ float fcos(float x) {
  return __builtin_amdgcn_cosf(x * 0.15915494309189535f);
}
__device__ __forceinline__ float ftanh(float x) {
  // tanh(x) = sign(x) * (1 - e^{-2|x|}) / (1 + e^{-2|x|})  (overflow-safe)
  float t = __builtin_amdgcn_exp2f(-2.885390081777927f * fabsf(x));
  float r = (1.0f - t) * __builtin_amdgcn_rcpf(1.0f + t);
  return copysignf(r, x);
}

__launch_bounds__(256)
__global__ void setdsr_kernel(const float* __restrict__ X,
                              const float* __restrict__ mask,
                              float* __restrict__ out, int N) {
  const int b   = blockIdx.x;
  const int tid = threadIdx.x;
  const float4* __restrict__ X4  = (const float4*)(X + (size_t)b * (size_t)N * 8);
  const float*  __restrict__ Mrw = mask + (size_t)b * (size_t)N;

  float s1 = 0.f, s2n = 0.f, s2d = 0.f, s4n = 0.f, s4d = 0.f;
  float lseM = -3.0e38f, lseS = 0.f, mx = NEG_INFF;

  // One pass over this row's 1024 elements (8 f32 features + 1 mask each).
  #pragma unroll 4
  for (int n = tid; n < N; n += 256) {
    float4 a = X4[2 * n + 0];        // x0..x3  (global_load_b128)
    float4 c = X4[2 * n + 1];        // x4..x7
    float  m = Mrw[n];
    bool valid = m > 0.f;

    // t1: sum(m * (x0*x1)^2)
    float p = a.x * a.y;
    s1 = fmaf(m, p * p, s1);

    // t2: sum(m * (safe_log(x2) + tanh(x3))), count = sum(m)
    float l = flog(fabsf(a.z) + EPSF) + ftanh(a.w);
    s2n = fmaf(m, l, s2n);
    s2d += m;

    // t3: online masked logsumexp of (x4 - x5)
    float e  = valid ? (c.x - c.y) : NEG_INFF;
    float Mn = fmaxf(lseM, e);
    lseS = fmaf(lseS, fexp(lseM - Mn), fexp(e - Mn));
    lseM = Mn;

    // t4: weighted mean, w = |x7| * m, v = x6
    float w = fabsf(c.w) * m;
    s4n = fmaf(c.z, w, s4n);
    s4d += w;

    // t5: masked max of sin(x0)*cos(x1)
    float sc = fsin(a.x) * fcos(a.y);
    mx = fmaxf(mx, valid ? sc : NEG_INFF);
  }

  // ---- intra-wave reduction (wave32) ----
  #pragma unroll
  for (int off = 16; off > 0; off >>= 1) {
    s1  += __shfl_xor(s1,  off, 32);
    s2n += __shfl_xor(s2n, off, 32);
    s2d += __shfl_xor(s2d, off, 32);
    s4n += __shfl_xor(s4n, off, 32);
    s4d += __shfl_xor(s4d, off, 32);
    float M2 = __shfl_xor(lseM, off, 32);
    float S2 = __shfl_xor(lseS, off, 32);
    float Mn = fmaxf(lseM, M2);
    lseS = lseS * fexp(lseM - Mn) + S2 * fexp(M2 - Mn);
    lseM = Mn;
    mx = fmaxf(mx, __shfl_xor(mx, off, 32));
  }

  // ---- cross-wave stage via LDS ----
  __shared__ float sred[8][16];   // 5 sum quantities, zero-padded to 16 cols
  __shared__ float lred[8][3];    // lseM, lseS, mx
  const int wave = tid >> 5;
  const int lane = tid & 31;
  if (lane == 0) {
    sred[wave][0] = s1;  sred[wave][1] = s2n; sred[wave][2] = s2d;
    sred[wave][3] = s4n; sred[wave][4] = s4d;
    #pragma unroll
    for (int q = 5; q < 16; ++q) sred[wave][q] = 0.f;
    lred[wave][0] = lseM; lred[wave][1] = lseS; lred[wave][2] = mx;
  }
  __syncthreads();

  // All 8 waves run the final reduction redundantly (no divergence -> EXEC is
  // all-ones for the WMMA, as the ISA requires); only tid 0 stores.
  //
  // Column-sum of the 8x16 partial table via two ones-matmuls:
  //   C[m][n] = sum_k A[m][k] * B[k][n], A = ones(16x4)  =>  C[*][n] = col n total.
  // B layout (16x16x4 f32): VGPR0 = B[lane>>4][lane&15], VGPR1 = B[2+(lane>>4)][lane&15].
  const int col  = lane & 15;
  const int rsel = lane >> 4;
  v2f A;  A[0]  = 1.0f;                A[1]  = 1.0f;
  v2f B1; B1[0] = sred[rsel + 0][col]; B1[1] = sred[rsel + 2][col];
  v2f B2; B2[0] = sred[rsel + 4][col]; B2[1] = sred[rsel + 6][col];
  v8f C = {};
  C = __builtin_amdgcn_wmma_f32_16x16x4_f32(false, A, false, B1, (short)0, C, false, false);
  C = __builtin_amdgcn_wmma_f32_16x16x4_f32(false, A, false, B2, (short)0, C, false, false);
  float tot = C[0];   // every lane: total of its column (quantity) index

  // LSE-merge + max across the 8 wave entries (xor tree over 8 lanes).
  const int w8 = lane & 7;
  float Mf = lred[w8][0], Sf = lred[w8][1], mxf = lred[w8][2];
  #pragma unroll
  for (int off = 4; off > 0; off >>= 1) {
    float M2 = __shfl_xor(Mf, off, 32);
    float S2 = __shfl_xor(Sf, off, 32);
    float Mn = fmaxf(Mf, M2);
    Sf = Sf * fexp(Mf - Mn) + S2 * fexp(M2 - Mn);
    Mf = Mn;
    mxf = fmaxf(mxf, __shfl_xor(mxf, off, 32));
  }

  float q0 = __shfl(tot, 0, 32);
  float q1 = __shfl(tot, 1, 32);
  float q2 = __shfl(tot, 2, 32);
  float q3 = __shfl(tot, 3, 32);
  float q4 = __shfl(tot, 4, 32);

  if (tid == 0) {
    float r = q0
            + q1 / (q2 + EPSF)
            + (Mf + flog(Sf))
            + q3 / (q4 + EPSF)
            + mxf;
    out[b] = r;
  }
}

extern "C" void kernel_launch(void* const* d_in, const int* in_sizes, int n_in,
                              void* d_out, int out_size, void* d_ws, size_t ws_size,
                              hipStream_t stream) {
  const float* X    = (const float*)d_in[0];
  const float* mask = (const float*)d_in[1];
  float* out = (float*)d_out;
  const int B = out_size;                 // 8192 rows, one block per row
  const int N = in_sizes[1] / B;          // 1024 set elements per row
  setdsr_kernel<<<dim3(B), dim3(256), 0, stream>>>(X, mask, out, N);
}